// AdjWeightedInverseDistance_9251359556316
// MI455X (gfx1250) — compile-verified
//
#include <hip/hip_runtime.h>
#include <hip/hip_bf16.h>

#define EPSV 1e-6f
#define NPTS 8192

typedef float v2f __attribute__((ext_vector_type(2)));
typedef float v8f __attribute__((ext_vector_type(8)));

// Workspace layout (floats):
//   [0      , 4*NPTS)  xs padded to float4  (x/lengthscale, w=0)
//   [4*NPTS , 5*NPTS)  sq[i]  = |xs_i|^2
//   [5*NPTS , 6*NPTS)  yc[i]  = y[i]*c[i]
//   [6*NPTS , 7*NPTS)  num[i] accumulator
//   [7*NPTS , 8*NPTS)  den[i] accumulator

__global__ void idw_prep_kernel(const float* __restrict__ x,
                                const float* __restrict__ c,
                                const float* __restrict__ y,
                                const float* __restrict__ ls,
                                float* __restrict__ ws) {
  int i = blockIdx.x * blockDim.x + threadIdx.x;
  if (i >= NPTS) return;
  float l0 = ls[0], l1 = ls[1], l2 = ls[2];
  float x0 = x[3 * i + 0] / l0;
  float x1 = x[3 * i + 1] / l1;
  float x2 = x[3 * i + 2] / l2;
  float4* xs4 = (float4*)ws;
  xs4[i] = make_float4(x0, x1, x2, 0.0f);
  ws[4 * NPTS + i] = x0 * x0 + x1 * x1 + x2 * x2;
  ws[5 * NPTS + i] = y[i] * c[i];
  ws[6 * NPTS + i] = 0.0f;   // num
  ws[7 * NPTS + i] = 0.0f;   // den
}

// One wave32 per block. blockIdx.x = 16-row tile, blockIdx.y = column chunk.
// Dot-products via V_WMMA_F32_16X16X4_F32 (D=3 padded to K=4 with zeros).
__global__ void __launch_bounds__(32)
idw_gram_kernel(const float* __restrict__ ws,
                const float* __restrict__ c,
                const float* __restrict__ powp,
                float* __restrict__ Kout,
                float* __restrict__ num,
                float* __restrict__ den,
                int jtiles_per_chunk) {
  const int lane = threadIdx.x & 31;
  const bool hi  = lane >= 16;
  const int  l   = lane & 15;
  const int  i0  = blockIdx.x * 16;
  const float p  = powp[0];

  const float4* xs4 = (const float4*)ws;
  const float*  sq  = ws + 4 * NPTS;
  const float*  yc  = ws + 5 * NPTS;

  // A fragment: rows i0..i0+15 of xs. Lane L<16 holds (M=L, K=0,1);
  // lane 16+L holds (M=L, K=2,3).  K=3 component is the zero pad.
  float4 xa = xs4[i0 + l];
  v2f a;
  a.x = hi ? xa.z : xa.x;
  a.y = hi ? xa.w : xa.y;

  // Per-row |xs_i|^2 for the 8 C-rows this lane's accumulator touches.
  float sqi[8];
#pragma unroll
  for (int v = 0; v < 8; ++v) sqi[v] = sq[i0 + v + (hi ? 8 : 0)];

  float accn[8] = {0.f, 0.f, 0.f, 0.f, 0.f, 0.f, 0.f, 0.f};
  float accd[8] = {0.f, 0.f, 0.f, 0.f, 0.f, 0.f, 0.f, 0.f};

  const int jt0 = blockIdx.y * jtiles_per_chunk;
  for (int jt = jt0; jt < jt0 + jtiles_per_chunk; ++jt) {
    const int j0  = jt * 16;
    const int col = j0 + l;

    // B fragment mirrors A (B rows striped across lanes like C/D).
    float4 xb = xs4[col];
    v2f b;
    b.x = hi ? xb.z : xb.x;
    b.y = hi ? xb.w : xb.y;

    const float sqj = sq[col];
    const float ycj = yc[col];
    const float cj  = c[col];

    v8f acc = {};
    acc = __builtin_amdgcn_wmma_f32_16x16x4_f32(
        /*neg_a=*/false, a, /*neg_b=*/false, b,
        /*c_mod=*/(short)0, acc, /*reuse_a=*/false, /*reuse_b=*/false);

#pragma unroll
    for (int v = 0; v < 8; ++v) {
      const int row = i0 + v + (hi ? 8 : 0);
      float d2   = sqi[v] + sqj - 2.0f * acc[v];
      float dist = __builtin_amdgcn_sqrtf(fmaxf(d2, 0.0f));
      // (dist+eps)^(-p) == exp2(-p * log2(dist+eps)) : 2 transcendentals
      // v_log_f32 is log2; v_exp_f32 is exp2.
      float kv = __builtin_amdgcn_exp2f(-p * __builtin_amdgcn_logf(dist + EPSV));
      Kout[(size_t)row * NPTS + col] = kv;
      accn[v] += kv * ycj;
      accd[v] += kv * cj;
    }
  }

  // Reduce within each 16-lane half (columns of the C tile), then one
  // atomicAdd per row per chunk. Lanes 0 and 16 hold rows v and v+8.
#pragma unroll
  for (int v = 0; v < 8; ++v) {
    float n_ = accn[v];
    float d_ = accd[v];
#pragma unroll
    for (int m = 8; m >= 1; m >>= 1) {
      n_ += __shfl_xor(n_, m, 32);
      d_ += __shfl_xor(d_, m, 32);
    }
    if (l == 0) {
      const int row = i0 + v + (hi ? 8 : 0);
      atomicAdd(&num[row], n_);
      atomicAdd(&den[row], d_);
    }
  }
}

__global__ void idw_finalize_kernel(const float* __restrict__ num,
                                    const float* __restrict__ den,
                                    float* __restrict__ yout) {
  int i = blockIdx.x * blockDim.x + threadIdx.x;
  if (i < NPTS) yout[i] = num[i] / den[i];
}

extern "C" void kernel_launch(void* const* d_in, const int* in_sizes, int n_in,
                              void* d_out, int out_size, void* d_ws, size_t ws_size,
                              hipStream_t stream) {
  const float* x  = (const float*)d_in[0];   // [N,3]
  const float* c  = (const float*)d_in[1];   // [N,1]
  const float* y  = (const float*)d_in[2];   // [N,1]
  const float* ls = (const float*)d_in[3];   // [3]
  const float* pw = (const float*)d_in[4];   // [1]

  float* out = (float*)d_out;       // [0,N): y_int ; [N, N+N*N): K
  float* K   = out + NPTS;
  float* ws  = (float*)d_ws;
  float* num = ws + 6 * NPTS;
  float* den = ws + 7 * NPTS;

  idw_prep_kernel<<<(NPTS + 255) / 256, 256, 0, stream>>>(x, c, y, ls, ws);

  const int JCHUNKS = 16;                       // 512 cols per wave sweep
  const int jtiles_per_chunk = (NPTS / 16) / JCHUNKS;  // 32 tiles of 16
  dim3 grid(NPTS / 16, JCHUNKS);
  idw_gram_kernel<<<grid, 32, 0, stream>>>(ws, c, pw, K, num, den,
                                           jtiles_per_chunk);

  idw_finalize_kernel<<<(NPTS + 255) / 256, 256, 0, stream>>>(num, den, out);
}